// Module_34342558498952
// MI455X (gfx1250) — compile-verified
//
#include <hip/hip_runtime.h>

#define F_CNT   1024
#define IMG_H   192
#define IMG_W   192
#define NPIX    (IMG_H * IMG_W)
#define FX_C    200.0f
#define FY_C    200.0f
#define CX_C    96.0f
#define CY_C    96.0f
#define EPS_F   1e-8f

typedef float v2f __attribute__((ext_vector_type(2)));
typedef float v8f __attribute__((ext_vector_type(8)));

// workspace layout (floats):
//   [0]                      loss accumulator
//   [64 .. 64+8192)          packed A operand: 4 float2-pairs x 1024 faces
//                            pair0=(k0,k1) pair1=(k2,k3) pair2=(k4,k5) pair3=(0,0)
//   [64+8192 .. +1024)       q  (opac*valid), then r, g, b (1024 each)
#define WS_A2   64
#define WS_QRGB (64 + 8*F_CNT)

__global__ void init_kernel(float* __restrict__ ws) { ws[0] = 0.0f; }

__global__ __launch_bounds__(F_CNT) void prep_kernel(
    const float* __restrict__ vp, const int* __restrict__ faces,
    const float* __restrict__ rq, const float* __restrict__ lsc,
    const float* __restrict__ cols, const float* __restrict__ lop,
    float* __restrict__ ws)
{
    __shared__ float zsh[F_CNT];
    const int f = threadIdx.x;

    // ---- face coordinate frame ----
    const int ia = faces[f*3+0], ib = faces[f*3+1], ic = faces[f*3+2];
    float ax = vp[ia*3+0], ay = vp[ia*3+1], az = vp[ia*3+2];
    float bx = vp[ib*3+0], by = vp[ib*3+1], bz = vp[ib*3+2];
    float cx = vp[ic*3+0], cy = vp[ic*3+1], cz = vp[ic*3+2];
    float tx = (ax+bx+cx)*(1.0f/3.0f);
    float ty = (ay+by+cy)*(1.0f/3.0f);
    float tz = (az+bz+cz)*(1.0f/3.0f);
    float e1x = bx-ax, e1y = by-ay, e1z = bz-az;
    float e2x = cx-ax, e2y = cy-ay, e2z = cz-az;
    float il = 1.0f/(sqrtf(e1x*e1x+e1y*e1y+e1z*e1z)+EPS_F);
    float xx = e1x*il, xy = e1y*il, xz = e1z*il;
    float nx = e1y*e2z - e1z*e2y;
    float ny = e1z*e2x - e1x*e2z;
    float nz = e1x*e2y - e1y*e2x;
    float inl = 1.0f/(sqrtf(nx*nx+ny*ny+nz*nz)+EPS_F);
    float zx = nx*inl, zy = ny*inl, zz = nz*inl;
    float yx = zy*xz - zz*xy;
    float yy = zz*xx - zx*xz;
    float yz = zx*xy - zy*xx;
    float Fm[3][3] = {{xx,yx,zx},{xy,yy,zy},{xz,yz,zz}};   // columns x,y,z

    // ---- local gaussian rotation & scales ----
    float qw = rq[f*4+0], qx = rq[f*4+1], qy = rq[f*4+2], qz = rq[f*4+3];
    float iq = 1.0f/(sqrtf(qw*qw+qx*qx+qy*qy+qz*qz)+EPS_F);
    qw*=iq; qx*=iq; qy*=iq; qz*=iq;
    float Rm[3][3];
    Rm[0][0]=1.0f-2.0f*(qy*qy+qz*qz); Rm[0][1]=2.0f*(qx*qy-qw*qz); Rm[0][2]=2.0f*(qx*qz+qw*qy);
    Rm[1][0]=2.0f*(qx*qy+qw*qz); Rm[1][1]=1.0f-2.0f*(qx*qx+qz*qz); Rm[1][2]=2.0f*(qy*qz-qw*qx);
    Rm[2][0]=2.0f*(qx*qz-qw*qy); Rm[2][1]=2.0f*(qy*qz+qw*qx); Rm[2][2]=1.0f-2.0f*(qx*qx+qy*qy);
    float sc[3] = { expf(lsc[f*3+0]), expf(lsc[f*3+1]), expf(lsc[f*3+2]) };

    // G = face_rs @ (diag(s) R^T)
    float G[3][3];
    #pragma unroll
    for (int i = 0; i < 3; ++i)
        #pragma unroll
        for (int j = 0; j < 3; ++j)
            G[i][j] = Fm[i][0]*sc[0]*Rm[j][0] + Fm[i][1]*sc[1]*Rm[j][1] + Fm[i][2]*sc[2]*Rm[j][2];
    float gsv[3];
    #pragma unroll
    for (int j = 0; j < 3; ++j)
        gsv[j] = sqrtf(G[0][j]*G[0][j] + G[1][j]*G[1][j] + G[2][j]*G[2][j]);
    float Rg[3][3];
    #pragma unroll
    for (int i = 0; i < 3; ++i)
        #pragma unroll
        for (int j = 0; j < 3; ++j)
            Rg[i][j] = G[i][j] / (gsv[j] + EPS_F);

    // rot -> quat -> (renormalize) -> rot, faithful to reference
    float m00 = Rg[0][0], m11 = Rg[1][1], m22 = Rg[2][2];
    float pw_ = 0.5f*sqrtf(fmaxf(EPS_F, 1.0f+m00+m11+m22));
    float px_ = 0.5f*sqrtf(fmaxf(EPS_F, 1.0f+m00-m11-m22));
    float py_ = 0.5f*sqrtf(fmaxf(EPS_F, 1.0f-m00+m11-m22));
    float pz_ = 0.5f*sqrtf(fmaxf(EPS_F, 1.0f-m00-m11+m22));
    px_ = copysignf(px_, Rg[2][1]-Rg[1][2]);
    py_ = copysignf(py_, Rg[0][2]-Rg[2][0]);
    pz_ = copysignf(pz_, Rg[1][0]-Rg[0][1]);
    float iq2 = 1.0f/(sqrtf(pw_*pw_+px_*px_+py_*py_+pz_*pz_)+EPS_F);
    pw_*=iq2; px_*=iq2; py_*=iq2; pz_*=iq2;
    float Rn[3][3];
    Rn[0][0]=1.0f-2.0f*(py_*py_+pz_*pz_); Rn[0][1]=2.0f*(px_*py_-pw_*pz_); Rn[0][2]=2.0f*(px_*pz_+pw_*py_);
    Rn[1][0]=2.0f*(px_*py_+pw_*pz_); Rn[1][1]=1.0f-2.0f*(px_*px_+pz_*pz_); Rn[1][2]=2.0f*(py_*pz_-pw_*px_);
    Rn[2][0]=2.0f*(px_*pz_-pw_*py_); Rn[2][1]=2.0f*(py_*pz_+pw_*px_); Rn[2][2]=1.0f-2.0f*(px_*px_+py_*py_);

    float cov[3][3];
    #pragma unroll
    for (int i = 0; i < 3; ++i)
        #pragma unroll
        for (int j = 0; j < 3; ++j)
            cov[i][j] = Rn[i][0]*gsv[0]*gsv[0]*Rn[j][0] + Rn[i][1]*gsv[1]*gsv[1]*Rn[j][1]
                      + Rn[i][2]*gsv[2]*gsv[2]*Rn[j][2];

    // ---- projection ----
    float X = tx, Y = ty, Z = tz;
    float validf = (Z > 0.2f) ? 1.0f : 0.0f;
    float Zc = fmaxf(Z, 0.2f);
    float mx = FX_C*X/Zc + CX_C;
    float my = FY_C*Y/Zc + CY_C;
    float j00 = FX_C/Zc, j02 = -FX_C*X/(Zc*Zc);
    float j11 = FY_C/Zc, j12 = -FY_C*Y/(Zc*Zc);
    float a2 = j00*j00*cov[0][0] + 2.0f*j00*j02*cov[0][2] + j02*j02*cov[2][2] + 0.3f;
    float b2 = j00*j11*cov[0][1] + j00*j12*cov[0][2] + j02*j11*cov[2][1] + j02*j12*cov[2][2];
    float c2 = j11*j11*cov[1][1] + 2.0f*j11*j12*cov[1][2] + j12*j12*cov[2][2] + 0.3f;
    float det = fmaxf(a2*c2 - b2*b2, EPS_F);
    float idet = 1.0f/det;
    float conA = c2*idet, conB = -b2*idet, conC = a2*idet;
    float qo = expf(lop[f]) * validf;

    // power = k0 + k1*px + k2*py + k3*px^2 + k4*py^2 + k5*px*py
    float k0 = -0.5f*(conA*mx*mx + conC*my*my) - conB*mx*my;
    float k1 = conA*mx + conB*my;
    float k2 = conC*my + conB*mx;
    float k3 = -0.5f*conA;
    float k4 = -0.5f*conC;
    float k5 = -conB;

    // ---- stable counting sort by Zc ----
    zsh[f] = Zc;
    __syncthreads();
    int rank = 0;
    for (int g = 0; g < F_CNT; ++g) {
        float zg = zsh[g];
        rank += (zg < Zc) || ((zg == Zc) && (g < f));
    }

    // packed, zero-padded A operand: branchless b64 loads in render kernel
    float2* A2 = (float2*)(ws + WS_A2);
    A2[0*F_CNT + rank] = make_float2(k0, k1);
    A2[1*F_CNT + rank] = make_float2(k2, k3);
    A2[2*F_CNT + rank] = make_float2(k4, k5);
    A2[3*F_CNT + rank] = make_float2(0.0f, 0.0f);

    float* qarr = ws + WS_QRGB;
    qarr[rank]           = qo;
    qarr[rank + 1*F_CNT] = cols[f*3+0];
    qarr[rank + 2*F_CNT] = cols[f*3+1];
    qarr[rank + 3*F_CNT] = cols[f*3+2];
}

__global__ __launch_bounds__(256) void render_kernel(
    const float* __restrict__ img_in, const float* __restrict__ mask,
    float* __restrict__ out, float* __restrict__ ws)
{
    const v2f*   A2 = (const v2f*)(ws + WS_A2);
    const float* qp = ws + WS_QRGB;
    const float* rp = qp + 1*F_CNT;
    const float* gp = qp + 2*F_CNT;
    const float* bp = qp + 3*F_CNT;

    __shared__ float pwsh[2][16 * 256];       // double-buffered power tiles (32 KB)
    __shared__ float sQ[2][16], sR[2][16], sG[2][16], sB[2][16];

    const int tid   = threadIdx.x;
    const int lane  = tid & 31;               // wave32
    const int n     = lane & 15;              // WMMA column / A-row index
    const int half  = lane >> 4;              // 0: K-rows {0,1}/{4,5}; 1: K-rows {2,3}/{6,7}
    const bool lower = (half == 0);
    const int rowOff = half * 8;              // D-tile: VGPR v holds M=v (+8 for upper lanes)
    const int blockBase = blockIdx.x * 256;
    const int waveBase  = tid & ~31;

    const int pixA = waveBase + n;            // tile 0 column pixel (block-relative)
    const int pixB = waveBase + 16 + n;       // tile 1
    const int gA = blockBase + pixA, gB = blockBase + pixB;
    const float pxA = (float)(gA % IMG_W), pyA = (float)(gA / IMG_W);
    const float pxB = (float)(gB % IMG_W), pyB = (float)(gB / IMG_W);

    // B operands: VGPR0 = rows K0 (lanes<16) / K2 (lanes>=16); VGPR1 = rows K1 / K3.
    // basis K-halves: [1, px, py, px^2] then [py^2, px*py, 0, 0]
    const v2f bA0 = { lower ? 1.0f    : pyA,   lower ? pxA     : pxA*pxA };
    const v2f bA1 = { lower ? pyA*pyA : 0.0f,  lower ? pxA*pyA : 0.0f    };
    const v2f bB0 = { lower ? 1.0f    : pyB,   lower ? pxB     : pxB*pxB };
    const v2f bB1 = { lower ? pyB*pyB : 0.0f,  lower ? pxB*pyB : 0.0f    };

    float T = 1.0f, accR = 0.0f, accG = 0.0f, accB = 0.0f, sw = 0.0f;

    // compute chunk -> buffer: 2 unconditional b64 A loads, 4 WMMAs, 16 LDS stores
    auto compute = [&](int chunk, int buf) {
        const int face = chunk*16 + n;
        const v2f aLo = A2[half*F_CNT + face];        // pair0 or pair1
        const v2f aHi = A2[(2 + half)*F_CNT + face];  // pair2 or zero pair3
        v8f acc0 = {};
        acc0 = __builtin_amdgcn_wmma_f32_16x16x4_f32(false, aLo, false, bA0, (short)0, acc0, false, false);
        acc0 = __builtin_amdgcn_wmma_f32_16x16x4_f32(false, aHi, false, bA1, (short)0, acc0, false, false);
        v8f acc1 = {};
        acc1 = __builtin_amdgcn_wmma_f32_16x16x4_f32(false, aLo, false, bB0, (short)0, acc1, false, false);
        acc1 = __builtin_amdgcn_wmma_f32_16x16x4_f32(false, aHi, false, bB1, (short)0, acc1, false, false);
        #pragma unroll
        for (int v = 0; v < 8; ++v) {
            pwsh[buf][(v + rowOff)*256 + pixA] = acc0[v];
            pwsh[buf][(v + rowOff)*256 + pixB] = acc1[v];
        }
        if (tid < 16) {
            const int fi = chunk*16 + tid;
            sQ[buf][tid] = qp[fi]; sR[buf][tid] = rp[fi];
            sG[buf][tid] = gp[fi]; sB[buf][tid] = bp[fi];
        }
    };

    // sequential front-to-back blend for this thread's pixel
    auto blend = [&](int buf) {
        #pragma unroll
        for (int i = 0; i < 16; ++i) {
            float pw = pwsh[buf][i*256 + tid];
            float al = fminf(sQ[buf][i] * __expf(fminf(pw, 0.0f)), 0.99f);
            float w  = al * T;
            accR += w * sR[buf][i]; accG += w * sG[buf][i]; accB += w * sB[buf][i];
            sw   += w;
            T    *= (1.0f - al);
        }
    };

    // software pipeline: compute(k+1) overlaps blend(k); one barrier per chunk
    compute(0, 0);
    __syncthreads();
    for (int k = 0; k < F_CNT/16 - 1; ++k) {
        compute(k + 1, (k + 1) & 1);
        blend(k & 1);
        __syncthreads();
    }
    blend((F_CNT/16 - 1) & 1);

    const int p = blockBase + tid;
    const float bg = 1.0f - sw;
    const float o0 = accR + bg, o1 = accG + bg, o2 = accB + bg;
    out[p]          = o0;
    out[NPIX + p]   = o1;
    out[2*NPIX + p] = o2;

    // loss partial: (rendered - (1 - mask + img*mask))^2
    const float m = mask[p];
    const float d0 = o0 - (1.0f - m + img_in[p]*m);
    const float d1 = o1 - (1.0f - m + img_in[NPIX + p]*m);
    const float d2 = o2 - (1.0f - m + img_in[2*NPIX + p]*m);
    float e = d0*d0 + d1*d1 + d2*d2;

    __syncthreads();
    pwsh[0][tid] = e;
    __syncthreads();
    for (int s = 128; s > 0; s >>= 1) {
        if (tid < s) pwsh[0][tid] += pwsh[0][tid + s];
        __syncthreads();
    }
    if (tid == 0) atomicAdd(&ws[0], pwsh[0][0]);
}

__global__ void finalize_kernel(const float* __restrict__ ws, float* __restrict__ out) {
    out[3*NPIX] = ws[0] * (1.0f / (3.0f * (float)NPIX));
}

extern "C" void kernel_launch(void* const* d_in, const int* in_sizes, int n_in,
                              void* d_out, int out_size, void* d_ws, size_t ws_size,
                              hipStream_t stream) {
    const float* img   = (const float*)d_in[0];   // (3,192,192)
    const float* mask  = (const float*)d_in[1];   // (192,192)
    const float* vp    = (const float*)d_in[2];   // (512,3)
    const int*   faces = (const int*)  d_in[3];   // (1024,3)
    const float* rq    = (const float*)d_in[4];   // (1024,4)
    const float* lsc   = (const float*)d_in[5];   // (1024,3)
    const float* cols  = (const float*)d_in[6];   // (1024,3)
    const float* lop   = (const float*)d_in[7];   // (1024,1)
    float* out = (float*)d_out;                   // 3*192*192 rendered + 1 loss
    float* ws  = (float*)d_ws;

    init_kernel<<<1, 1, 0, stream>>>(ws);
    prep_kernel<<<1, F_CNT, 0, stream>>>(vp, faces, rq, lsc, cols, lop, ws);
    render_kernel<<<NPIX/256, 256, 0, stream>>>(img, mask, out, ws);
    finalize_kernel<<<1, 1, 0, stream>>>(ws, out);
}